// SparseResidualBlock_8581344657790
// MI455X (gfx1250) — compile-verified
//
#include <hip/hip_runtime.h>

typedef float v2f __attribute__((ext_vector_type(2)));
typedef float v8f __attribute__((ext_vector_type(8)));

#define TILE_ROWS 32
#define KC 576            // K*C = 9*64 (GEMM reduction length)
#define LDS_STRIDE 580    // 576 + 4 dwords padding -> conflict-free ds_load_b64
#define KSTEPS 144        // KC / 4 (one v_wmma_f32_16x16x4_f32 per step)

// ---------------------------------------------------------------------------
// Swizzle W [9,64,64] (== row-major [576,64]) into per-lane B-fragment order:
// for col-tile t (0..3), k-step s (0..143), lane L (0..31):
//   k0 = 4*s + 2*(L>=16), n = t*16 + (L&15),  out[idx] = {Wf[k0][n], Wf[k0+1][n]}
// so the GEMM inner loop does one coalesced 8-byte load per lane.
// ---------------------------------------------------------------------------
__global__ __launch_bounds__(256) void swz_weights(const float* __restrict__ W,
                                                   float* __restrict__ Bswz) {
  int idx = blockIdx.x * 256 + threadIdx.x;
  if (idx >= 4 * KSTEPS * 32) return;
  int L = idx & 31;
  int s = (idx >> 5) % KSTEPS;
  int t = idx / (KSTEPS * 32);
  int k0 = 4 * s + ((L >> 4) << 1);
  int n  = t * 16 + (L & 15);
  v2f v;
  v.x = W[k0 * 64 + n];
  v.y = W[(k0 + 1) * 64 + n];
  reinterpret_cast<v2f*>(Bswz)[idx] = v;
}

// ---------------------------------------------------------------------------
// One fused subm-conv pass: gather (optionally applying previous BN affine +
// ReLU per channel) -> LDS, then [32 x 576] x [576 x 64] via f32 WMMA with
// dual accumulators (breaks the D->C RAW chain), store D and per-block
// channel sum / sum-of-squares (fixed-order, no atomics -> deterministic).
// ---------------------------------------------------------------------------
__global__ __launch_bounds__(256) void subm_conv_wmma(
    const float* __restrict__ src, const int* __restrict__ nbrs,
    const float* __restrict__ Bswz,
    const float* __restrict__ affS, const float* __restrict__ affB, int doAffine,
    float* __restrict__ dst, float* __restrict__ partS, float* __restrict__ partQ,
    int N) {
  __shared__ float Alds[TILE_ROWS * LDS_STRIDE];
  __shared__ float statS[128];
  __shared__ float statQ[128];

  const int tid = threadIdx.x;
  const int r0  = blockIdx.x * TILE_ROWS;

  // ---- gather stage: 288 source rows (32 sites x 9 taps), 64 ch each ----
  {
    const int cg = tid & 15;   // channel group: handles channels cg*4 .. cg*4+3
    const int rg = tid >> 4;   // gather-row within a 16-row chunk
    float sA[4] = {1.f, 1.f, 1.f, 1.f};
    float bA[4] = {0.f, 0.f, 0.f, 0.f};
    if (doAffine) {
#pragma unroll
      for (int j = 0; j < 4; ++j) {
        sA[j] = affS[cg * 4 + j];
        bA[j] = affB[cg * 4 + j];
      }
    }
    const float4* src4 = reinterpret_cast<const float4*>(src);
#pragma unroll
    for (int it = 0; it < 18; ++it) {
      int g = it * 16 + rg;                 // 0..287
      int m = g / 9;
      int k = g - m * 9;
      int site = r0 + m;
      if (site >= N) site = N - 1;
      int srow = nbrs[site * 9 + k];        // random, but L2-resident table
      float4 v = src4[srow * 16 + cg];      // global_load_b128, coalesced
      if (doAffine) {                       // fused BN1 + ReLU on gathered vals
        v.x = fmaxf(fmaf(v.x, sA[0], bA[0]), 0.f);
        v.y = fmaxf(fmaf(v.y, sA[1], bA[1]), 0.f);
        v.z = fmaxf(fmaf(v.z, sA[2], bA[2]), 0.f);
        v.w = fmaxf(fmaf(v.w, sA[3], bA[3]), 0.f);
      }
      *reinterpret_cast<float4*>(&Alds[m * LDS_STRIDE + k * 64 + cg * 4]) = v;
    }
  }
  __syncthreads();

  // ---- WMMA stage: each wave owns a 16x16 D tile ----
  const int wave = tid >> 5;
  const int lane = tid & 31;
  const int trow = wave >> 2;          // 0..1  (row tile)
  const int tcol = wave & 3;           // 0..3  (col tile)
  const int m    = (lane & 15) + trow * 16;
  const int kb   = (lane >> 4) << 1;   // A frag: VGPR0=K0/K2, VGPR1=K1/K3

  const v2f*   Bp = reinterpret_cast<const v2f*>(Bswz) + (tcol * KSTEPS) * 32 + lane;
  const float* Ap = &Alds[m * LDS_STRIDE + kb];

  v8f acc0 = {};
  v8f acc1 = {};
#pragma unroll 4
  for (int s = 0; s < KSTEPS; s += 2) {
    v2f a0 = *reinterpret_cast<const v2f*>(Ap + 4 * s);        // ds_load_b64
    v2f b0 = Bp[s * 32];                                       // global_load_b64
    v2f a1 = *reinterpret_cast<const v2f*>(Ap + 4 * s + 4);
    v2f b1 = Bp[(s + 1) * 32];
    acc0 = __builtin_amdgcn_wmma_f32_16x16x4_f32(false, a0, false, b0, (short)0,
                                                 acc0, false, false);
    acc1 = __builtin_amdgcn_wmma_f32_16x16x4_f32(false, a1, false, b1, (short)0,
                                                 acc1, false, false);
  }
  v8f acc = acc0 + acc1;

  // ---- store D tile (C/D layout: VGPR i -> M=i (lanes 0-15) / M=8+i) ----
  const int col  = tcol * 16 + (lane & 15);
  const int row0 = r0 + trow * 16 + ((lane >> 4) << 3);
#pragma unroll
  for (int i = 0; i < 8; ++i) {
    int row = row0 + i;
    if (row < N) dst[row * 64 + col] = acc[i];
  }

  // ---- per-block channel statistics (deterministic fixed-order) ----
  float sum = 0.f, sq = 0.f;
#pragma unroll
  for (int i = 0; i < 8; ++i) {
    float v = acc[i];
    sum += v;
    sq  += v * v;
  }
  sum += __shfl_xor(sum, 16, 32);  // combine the two 8-row halves (same col)
  sq  += __shfl_xor(sq, 16, 32);
  if (lane < 16) {
    statS[wave * 16 + lane] = sum;
    statQ[wave * 16 + lane] = sq;
  }
  __syncthreads();
  if (tid < 64) {  // channel ch: waves tcol=ch/16 at trow 0 and 1
    partS[blockIdx.x * 64 + tid] = statS[tid] + statS[64 + tid];
    partQ[blockIdx.x * 64 + tid] = statQ[tid] + statQ[64 + tid];
  }
}

// ---------------------------------------------------------------------------
// Fold per-block partials into the fused BN affine: s = gamma*rsqrt(var+eps),
// b = beta - mean*s.  64 threads, fixed summation order -> deterministic.
// ---------------------------------------------------------------------------
__global__ void finalize_stats(const float* __restrict__ pS,
                               const float* __restrict__ pQ, int nBlocks,
                               const float* __restrict__ gamma,
                               const float* __restrict__ beta, float invN,
                               float* __restrict__ sOut, float* __restrict__ bOut) {
  int c = threadIdx.x;  // 64 threads
  float s = 0.f, q = 0.f;
  for (int b = 0; b < nBlocks; ++b) {
    s += pS[(size_t)b * 64 + c];
    q += pQ[(size_t)b * 64 + c];
  }
  float mean = s * invN;
  float var  = q * invN - mean * mean;
  float rstd = rsqrtf(var + 1e-5f);
  float sc   = rstd * gamma[c];
  sOut[c] = sc;
  bOut[c] = beta[c] - mean * sc;
}

// out = relu(BN2(conv2) + x), conv2 result already sitting in `out`
__global__ void epilogue(float* __restrict__ out, const float* __restrict__ x,
                         const float* __restrict__ s2, const float* __restrict__ b2,
                         int total) {
  int i = blockIdx.x * blockDim.x + threadIdx.x;
  if (i < total) {
    int c = i & 63;
    out[i] = fmaxf(fmaf(out[i], s2[c], b2[c]) + x[i], 0.f);
  }
}

extern "C" void kernel_launch(void* const* d_in, const int* in_sizes, int n_in,
                              void* d_out, int out_size, void* d_ws, size_t ws_size,
                              hipStream_t stream) {
  (void)n_in; (void)out_size; (void)ws_size;
  const float* x   = (const float*)d_in[0];
  const int*   nb  = (const int*)d_in[1];
  const float* W1  = (const float*)d_in[2];
  const float* g1  = (const float*)d_in[3];
  const float* be1 = (const float*)d_in[4];
  const float* W2  = (const float*)d_in[5];
  const float* g2  = (const float*)d_in[6];
  const float* be2 = (const float*)d_in[7];

  const int N = in_sizes[0] / 64;
  const int nBlocks = (N + TILE_ROWS - 1) / TILE_ROWS;

  float* ws    = (float*)d_ws;
  float* Bswz1 = ws;                                   // 36864 floats
  float* Bswz2 = Bswz1 + 36864;                        // 36864
  float* s1    = Bswz2 + 36864;                        // 64
  float* b1    = s1 + 64;                              // 64
  float* s2    = b1 + 64;                              // 64
  float* b2    = s2 + 64;                              // 64
  float* pS1   = b2 + 64;                              // nBlocks*64
  float* pQ1   = pS1 + (size_t)nBlocks * 64;
  float* pS2   = pQ1 + (size_t)nBlocks * 64;
  float* pQ2   = pS2 + (size_t)nBlocks * 64;
  float* tmp1  = pQ2 + (size_t)nBlocks * 64;           // N*64 raw conv1 output
  float* out   = (float*)d_out;

  const int swzElems = 4 * KSTEPS * 32;
  swz_weights<<<(swzElems + 255) / 256, 256, 0, stream>>>(W1, Bswz1);
  swz_weights<<<(swzElems + 255) / 256, 256, 0, stream>>>(W2, Bswz2);

  // conv1: plain gather, raw output + stats
  subm_conv_wmma<<<nBlocks, 256, 0, stream>>>(x, nb, Bswz1, s1, b1, 0,
                                              tmp1, pS1, pQ1, N);
  finalize_stats<<<1, 64, 0, stream>>>(pS1, pQ1, nBlocks, g1, be1, 1.0f / N, s1, b1);

  // conv2: BN1+ReLU fused into the gather; result into d_out + stats
  subm_conv_wmma<<<nBlocks, 256, 0, stream>>>(tmp1, nb, Bswz2, s1, b1, 1,
                                              out, pS2, pQ2, N);
  finalize_stats<<<1, 64, 0, stream>>>(pS2, pQ2, nBlocks, g2, be2, 1.0f / N, s2, b2);

  // BN2 + residual + ReLU in place
  const int total = N * 64;
  epilogue<<<(total + 255) / 256, 256, 0, stream>>>(out, x, s2, b2, total);
}